// Decoder_2027224563764
// MI455X (gfx1250) — compile-verified
//
#include <hip/hip_runtime.h>
#include <math.h>

typedef __attribute__((ext_vector_type(2))) float v2f;
typedef __attribute__((ext_vector_type(8))) float v8f;

#define B_   64
#define L_   196
#define DE   2048
#define DH   512
#define V_   10000
#define T_   21
#define NSTEP 20

#define ACT_NONE 0
#define ACT_TANH 1
#define ACT_SIG  2

__device__ __forceinline__ float sigf(float x) { return 1.0f / (1.0f + expf(-x)); }

// ---------------------------------------------------------------------------
// Fused GEMM: C[M,N] = act( A[M,K] @ W[K,N] + bias[N] (+ C if accum) )
// A row-major (lda), W row-major (ldw), C row-major (ldc).
// M multiple of 64, N multiple of 16, K multiple of 4.
// One wave computes a 64x16 strip = 4 stacked 16x16 WMMA tiles that share the
// B fragment (register blocking: 4 v_wmma_f32_16x16x4_f32 per 6 VMEM ops).
// ---------------------------------------------------------------------------
__global__ void wmma_gemm_f32(const float* __restrict__ A, int lda,
                              const float* __restrict__ W, int ldw,
                              const float* __restrict__ bias,
                              float* __restrict__ C, int ldc,
                              int M, int N, int K, int act, int accum)
{
    const int lane = threadIdx.x & 31;
    const int wid  = blockIdx.x * (blockDim.x >> 5) + (threadIdx.x >> 5);
    const int tn_count = N >> 4;
    const int total    = (M >> 6) * tn_count;    // strips of 64 rows
    if (wid >= total) return;                    // wave-uniform exit

    const int tm = (wid / tn_count) << 6;
    const int tn = (wid % tn_count) << 4;

    // A 16x4 fp32 layout: lanes 0-15 -> rows M=0..15 (K pair 0,1),
    //                     lanes 16-31 -> rows M=0..15 (K pair 2,3)
    const int mrow = tm + (lane & 15);
    const int ncol = tn + (lane & 15);
    const int kb   = (lane >> 4) << 1;           // 0 or 2

    const float* a0   = A + (size_t)(mrow +  0) * lda + kb;
    const float* a1   = A + (size_t)(mrow + 16) * lda + kb;
    const float* a2   = A + (size_t)(mrow + 32) * lda + kb;
    const float* a3   = A + (size_t)(mrow + 48) * lda + kb;
    const float* wcol = W + (size_t)kb * ldw + ncol;

    v8f acc[4];
#pragma unroll
    for (int i = 0; i < 4; ++i) acc[i] = (v8f){0.f,0.f,0.f,0.f,0.f,0.f,0.f,0.f};

    for (int k = 0; k < K; k += 4) {
        v2f b;                                   // shared B fragment
        b.x = wcol[(size_t)k * ldw];
        b.y = wcol[(size_t)(k + 1) * ldw];
        v2f x0 = *(const v2f*)(a0 + k);          // 8B aligned: lda, kb, k even
        v2f x1 = *(const v2f*)(a1 + k);
        v2f x2 = *(const v2f*)(a2 + k);
        v2f x3 = *(const v2f*)(a3 + k);
        acc[0] = __builtin_amdgcn_wmma_f32_16x16x4_f32(false, x0, false, b, (short)0, acc[0], false, false);
        acc[1] = __builtin_amdgcn_wmma_f32_16x16x4_f32(false, x1, false, b, (short)0, acc[1], false, false);
        acc[2] = __builtin_amdgcn_wmma_f32_16x16x4_f32(false, x2, false, b, (short)0, acc[2], false, false);
        acc[3] = __builtin_amdgcn_wmma_f32_16x16x4_f32(false, x3, false, b, (short)0, acc[3], false, false);
    }

    // D layout: vgpr r, lanes 0-15 -> M = r ; lanes 16-31 -> M = r+8 ; N = lane&15
    const int r0 = (lane >> 4) << 3;
    const float bn = bias[ncol];
#pragma unroll
    for (int mt = 0; mt < 4; ++mt) {
#pragma unroll
        for (int r = 0; r < 8; ++r) {
            const int mm = tm + (mt << 4) + r0 + r;
            float v = acc[mt][r] + bn;
            if (accum) v += C[(size_t)mm * ldc + ncol];
            if (act == ACT_TANH)     v = tanhf(v);
            else if (act == ACT_SIG) v = sigf(v);
            C[(size_t)mm * ldc + ncol] = v;
        }
    }
}

// avg[b,d] = mean over L of img[b,l,d]
__global__ void avg_kernel(const float* __restrict__ img, float* __restrict__ avg)
{
    int idx = blockIdx.x * blockDim.x + threadIdx.x;
    if (idx >= B_ * DE) return;
    int b = idx / DE, d = idx % DE;
    const float* p = img + (size_t)b * L_ * DE + d;
    float s = 0.f;
    for (int l = 0; l < L_; ++l) s += p[(size_t)l * DE];
    avg[idx] = s * (1.0f / (float)L_);
}

// e[b,l] = v_b + sum_d tanh(Ws[b,l,d] + hU[b,d]) * v_w[d]   (one wave per (b,l))
__global__ void att_e_kernel(const float* __restrict__ Ws, const float* __restrict__ hU,
                             const float* __restrict__ v_w, const float* __restrict__ v_b,
                             float* __restrict__ e)
{
    int lane = threadIdx.x & 31;
    int idx  = blockIdx.x * (blockDim.x >> 5) + (threadIdx.x >> 5);
    if (idx >= B_ * L_) return;
    int b = idx / L_;
    const float* ws = Ws + (size_t)idx * DH;
    const float* hu = hU + (size_t)b * DH;
    float s = 0.f;
    for (int d = lane; d < DH; d += 32)
        s += tanhf(ws[d] + hu[d]) * v_w[d];
    for (int m = 16; m >= 1; m >>= 1) s += __shfl_xor(s, m, 32);
    if (lane == 0) e[idx] = s + v_b[0];
}

// softmax over L per batch row; also scatter alpha into output tensor at step t
__global__ void softmax_kernel(const float* __restrict__ e, float* __restrict__ alpha,
                               float* __restrict__ out_alpha, int t)
{
    __shared__ float red[256];
    int b = blockIdx.x, tid = threadIdx.x;
    float v = (tid < L_) ? e[b * L_ + tid] : -3.4e38f;
    red[tid] = v; __syncthreads();
    for (int s = 128; s >= 1; s >>= 1) {
        if (tid < s) red[tid] = fmaxf(red[tid], red[tid + s]);
        __syncthreads();
    }
    float mx = red[0]; __syncthreads();
    float ex = (tid < L_) ? expf(v - mx) : 0.f;
    red[tid] = ex; __syncthreads();
    for (int s = 128; s >= 1; s >>= 1) {
        if (tid < s) red[tid] += red[tid + s];
        __syncthreads();
    }
    float inv = 1.0f / red[0];
    if (tid < L_) {
        float a = ex * inv;
        alpha[b * L_ + tid] = a;
        out_alpha[((size_t)b * NSTEP + t) * L_ + tid] = a;
    }
}

// ctx[b,d] = sum_l alpha[b,l] * img[b,l,d]
__global__ void context_kernel(const float* __restrict__ img, const float* __restrict__ alpha,
                               float* __restrict__ ctx)
{
    __shared__ float sa[L_];
    int b = blockIdx.y;
    int d = blockIdx.x * blockDim.x + threadIdx.x;
    for (int l = threadIdx.x; l < L_; l += blockDim.x) sa[l] = alpha[b * L_ + l];
    __syncthreads();
    const float* p = img + (size_t)b * L_ * DE + d;
    float s = 0.f;
    for (int l = 0; l < L_; ++l) s = fmaf(sa[l], p[(size_t)l * DE], s);
    ctx[b * DE + d] = s;
}

// lstm_in[b, 0:DH]     = emb[captions[b,t]]
// lstm_in[b, DH:DH+DE] = gate[b] * ctx[b]
__global__ void lstm_in_kernel(const int* __restrict__ captions, const float* __restrict__ emb,
                               const float* __restrict__ gate, const float* __restrict__ ctx,
                               float* __restrict__ lstm_in, int t)
{
    int idx = blockIdx.x * blockDim.x + threadIdx.x;
    if (idx >= B_ * (DH + DE)) return;
    int b = idx / (DH + DE), j = idx % (DH + DE);
    float v;
    if (j < DH) {
        int tok = captions[b * T_ + t];
        v = emb[(size_t)tok * DH + j];
    } else {
        int d = j - DH;
        v = gate[b * DE + d] * ctx[b * DE + d];
    }
    lstm_in[idx] = v;
}

// LSTM cell pointwise: gates[b, 4*DH] in (i,f,g,o) order
__global__ void lstm_cell_kernel(const float* __restrict__ gates, const float* __restrict__ c_in,
                                 float* __restrict__ c_out, float* __restrict__ h_out)
{
    int idx = blockIdx.x * blockDim.x + threadIdx.x;
    if (idx >= B_ * DH) return;
    int b = idx / DH, j = idx % DH;
    const float* g = gates + (size_t)b * 4 * DH;
    float i  = sigf(g[j]);
    float f  = sigf(g[DH + j]);
    float gg = tanhf(g[2 * DH + j]);
    float o  = sigf(g[3 * DH + j]);
    float c  = f * c_in[idx] + i * gg;
    c_out[idx] = c;
    h_out[idx] = o * tanhf(c);
}

static inline void launch_gemm(const float* A, int lda, const float* W, int ldw,
                               const float* bias, float* C, int ldc,
                               int M, int N, int K, int act, int accum, hipStream_t s)
{
    int tiles  = (M >> 6) * (N >> 4);          // one wave per 64x16 strip
    int blocks = (tiles + 7) / 8;              // 8 waves (256 threads) per block
    wmma_gemm_f32<<<blocks, 256, 0, s>>>(A, lda, W, ldw, bias, C, ldc, M, N, K, act, accum);
}

extern "C" void kernel_launch(void* const* d_in, const int* in_sizes, int n_in,
                              void* d_out, int out_size, void* d_ws, size_t ws_size,
                              hipStream_t stream)
{
    const float* img       = (const float*)d_in[0];
    const int*   captions  = (const int*)  d_in[1];
    const float* U_w       = (const float*)d_in[2];
    const float* U_b       = (const float*)d_in[3];
    const float* Wm_w      = (const float*)d_in[4];
    const float* Wm_b      = (const float*)d_in[5];
    const float* v_w       = (const float*)d_in[6];
    const float* v_b       = (const float*)d_in[7];
    const float* init_h_w  = (const float*)d_in[8];
    const float* init_h_b  = (const float*)d_in[9];
    const float* init_c_w  = (const float*)d_in[10];
    const float* init_c_b  = (const float*)d_in[11];
    const float* f_beta_w  = (const float*)d_in[12];
    const float* f_beta_b  = (const float*)d_in[13];
    const float* deep_w    = (const float*)d_in[14];
    const float* deep_b    = (const float*)d_in[15];
    const float* emb       = (const float*)d_in[16];
    const float* W_ih      = (const float*)d_in[17];
    const float* b_ih      = (const float*)d_in[18];
    const float* W_hh      = (const float*)d_in[19];
    const float* b_hh      = (const float*)d_in[20];

    float* out_preds  = (float*)d_out;                               // [B, 20, V]
    float* out_alphas = out_preds + (size_t)B_ * NSTEP * V_;         // [B, 20, L]

    // Workspace layout (floats)
    float* w = (float*)d_ws;
    float* avg     = w;  w += B_ * DE;
    float* Ws      = w;  w += (size_t)B_ * L_ * DH;
    float* hU      = w;  w += B_ * DH;
    float* e       = w;  w += B_ * L_;
    float* alpha   = w;  w += B_ * L_;
    float* ctx     = w;  w += B_ * DE;
    float* gate    = w;  w += B_ * DE;
    float* lstm_in = w;  w += B_ * (DH + DE);
    float* gates   = w;  w += B_ * 4 * DH;
    float* hbuf0   = w;  w += B_ * DH;
    float* hbuf1   = w;  w += B_ * DH;
    float* cbuf0   = w;  w += B_ * DH;
    float* cbuf1   = w;  w += B_ * DH;
    (void)ws_size; (void)n_in; (void)in_sizes; (void)out_size;

    // ---- Precompute ----
    avg_kernel<<<(B_ * DE + 255) / 256, 256, 0, stream>>>(img, avg);
    launch_gemm(avg, DE, init_h_w, DH, init_h_b, hbuf0, DH, B_, DH, DE, ACT_TANH, 0, stream);
    launch_gemm(avg, DE, init_c_w, DH, init_c_b, cbuf0, DH, B_, DH, DE, ACT_TANH, 0, stream);
    launch_gemm(img, DE, Wm_w, DH, Wm_b, Ws, DH, B_ * L_, DH, DE, ACT_NONE, 0, stream);

    // ---- Sequential scan ----
    float* hptr[2] = { hbuf0, hbuf1 };
    float* cptr[2] = { cbuf0, cbuf1 };
    for (int t = 0; t < NSTEP; ++t) {
        float* hcur = hptr[t & 1];
        float* ccur = cptr[t & 1];
        float* hnxt = hptr[(t + 1) & 1];
        float* cnxt = cptr[(t + 1) & 1];

        launch_gemm(hcur, DH, U_w, DH, U_b, hU, DH, B_, DH, DH, ACT_NONE, 0, stream);
        att_e_kernel<<<(B_ * L_ + 7) / 8, 256, 0, stream>>>(Ws, hU, v_w, v_b, e);
        softmax_kernel<<<B_, 256, 0, stream>>>(e, alpha, out_alphas, t);
        context_kernel<<<dim3(DE / 256, B_), 256, 0, stream>>>(img, alpha, ctx);
        launch_gemm(hcur, DH, f_beta_w, DE, f_beta_b, gate, DE, B_, DE, DH, ACT_SIG, 0, stream);
        lstm_in_kernel<<<(B_ * (DH + DE) + 255) / 256, 256, 0, stream>>>(
            captions, emb, gate, ctx, lstm_in, t);
        launch_gemm(lstm_in, DH + DE, W_ih, 4 * DH, b_ih, gates, 4 * DH,
                    B_, 4 * DH, DH + DE, ACT_NONE, 0, stream);
        launch_gemm(hcur, DH, W_hh, 4 * DH, b_hh, gates, 4 * DH,
                    B_, 4 * DH, DH, ACT_NONE, 1, stream);
        lstm_cell_kernel<<<(B_ * DH + 255) / 256, 256, 0, stream>>>(gates, ccur, cnxt, hnxt);
        launch_gemm(hnxt, DH, deep_w, V_, deep_b, out_preds + (size_t)t * V_,
                    NSTEP * V_, B_, V_, DH, ACT_NONE, 0, stream);
    }
}